// ReLiKRelationExtractor_32074815766713
// MI455X (gfx1250) — compile-verified
//
#include <hip/hip_runtime.h>
#include <hip/hip_bf16.h>

typedef __attribute__((ext_vector_type(2))) float v2f;
typedef __attribute__((ext_vector_type(8))) float v8f;

// ---------------------------------------------------------------------------
// Stage 1: P[128,1024] = stacked GEMM, f32 WMMA 16x16x4.
//   rows   0..47  : S  = entity_emb @ W1[   0:1024, :]
//   rows  48..79  : RL = rel_emb    @ W1[1024:2048, :]
//   rows  80..127 : O  = entity_emb @ W1[2048:3072, :]
// One wave computes a 16(M) x 64(N) tile (4 WMMA accumulators, shared A frag).
// Block = 4 waves -> 16 x 256.  Grid = (8 Mtiles, 4 Nblocks).
// ---------------------------------------------------------------------------
__global__ __launch_bounds__(128) void relik_stage1_wmma(
    const float* __restrict__ ent, const float* __restrict__ rel,
    const float* __restrict__ W1, float* __restrict__ P) {
  constexpr int H = 1024;
  const int lane  = threadIdx.x & 31;
  const int wave  = threadIdx.x >> 5;
  const int mtile = blockIdx.x;                       // 0..7
  const int nbase = blockIdx.y * 256 + wave * 64;     // start of 4 N-tiles

  // Segment select (uniform within a 16-row M tile: 48 and 80 are 16-aligned)
  const int m0 = mtile * 16;
  const float* A;
  int arow0, seg;
  if (m0 < 48)      { A = ent; arow0 = m0;      seg = 0; }
  else if (m0 < 80) { A = rel; arow0 = m0 - 48; seg = 1; }
  else              { A = ent; arow0 = m0 - 80; seg = 2; }

  // A fragment (16x4 f32): lane l, elem v -> M = l&15, K = kb + 2*(l>>4) + v
  const float* __restrict__ Arow = A + (arow0 + (lane & 15)) * H;
  const int koff = (lane >> 4) * 2;

  // B fragment (4x16 f32): lane l, elem v -> K = kb + 2*(l>>4) + v, N = n + (l&15)
  const float* __restrict__ Bbase = W1 + (size_t)seg * H * H;  // row-major [H,H]
  const int bcol = lane & 15;

  v8f c0 = {}, c1 = {}, c2 = {}, c3 = {};
  for (int kb = 0; kb < H; kb += 4) {
    v2f a;
    a.x = Arow[kb + koff + 0];
    a.y = Arow[kb + koff + 1];
    const float* Br = Bbase + (size_t)(kb + koff) * H + bcol;
    v2f b0, b1v, b2v, b3v;
    b0.x  = Br[nbase +  0];  b0.y  = Br[H + nbase +  0];
    b1v.x = Br[nbase + 16];  b1v.y = Br[H + nbase + 16];
    b2v.x = Br[nbase + 32];  b2v.y = Br[H + nbase + 32];
    b3v.x = Br[nbase + 48];  b3v.y = Br[H + nbase + 48];
    c0 = __builtin_amdgcn_wmma_f32_16x16x4_f32(false, a, false, b0,  (short)0, c0, false, false);
    c1 = __builtin_amdgcn_wmma_f32_16x16x4_f32(false, a, false, b1v, (short)0, c1, false, false);
    c2 = __builtin_amdgcn_wmma_f32_16x16x4_f32(false, a, false, b2v, (short)0, c2, false, false);
    c3 = __builtin_amdgcn_wmma_f32_16x16x4_f32(false, a, false, b3v, (short)0, c3, false, false);
  }

  // D layout: VGPR v, lanes0-15 -> M=v, lanes16-31 -> M=v+8, N = lane&15
  float* __restrict__ outp = P + (size_t)(m0 + (lane >> 4) * 8) * H + nbase + bcol;
#pragma unroll
  for (int v = 0; v < 8; ++v) {
    outp[(size_t)v * H +  0] = c0[v];
    outp[(size_t)v * H + 16] = c1[v];
    outp[(size_t)v * H + 32] = c2[v];
    outp[(size_t)v * H + 48] = c3[v];
  }
}

// ---------------------------------------------------------------------------
// Stage 2: one block per (i,j) pair.  out[i,j,k] =
//   mask ? sigmoid( sum_h relu(S[i,h]+RL[k,h]+O[j,h]+b1[h]) * W2[h] + b2 ) : 0
// 256 threads = 8 waves; each wave owns 4 relations, reduces over H=1024.
// SO row and W2 staged in LDS (8 KB).  RL rows stream from L2-resident P.
// ---------------------------------------------------------------------------
__global__ __launch_bounds__(256) void relik_stage2(
    const float* __restrict__ P, const float* __restrict__ b1,
    const float* __restrict__ W2, const float* __restrict__ b2,
    const int* __restrict__ starts, const int* __restrict__ maxd,
    float* __restrict__ out) {
  constexpr int H = 1024, E = 48, R = 32;
  const int i = blockIdx.x, j = blockIdx.y;
  float* __restrict__ outp = out + ((size_t)i * E + j) * R;

  int d = starts[i] - starts[j];
  if (d < 0) d = -d;
  const bool masked = (i == j) || (d > maxd[0]);
  if (masked) {
    if (threadIdx.x < R) outp[threadIdx.x] = 0.0f;
    return;
  }

  __shared__ float sSO[H];
  __shared__ float sW2[H];
  const float* __restrict__ Si = P + (size_t)i * H;
  const float* __restrict__ Oj = P + (size_t)(80 + j) * H;
  for (int h = threadIdx.x; h < H; h += 256) {
    sSO[h] = Si[h] + Oj[h] + b1[h];
    sW2[h] = W2[h];
  }
  __syncthreads();

  const int lane = threadIdx.x & 31;
  const int wave = threadIdx.x >> 5;  // 0..7
  const float bias2 = b2[0];
#pragma unroll
  for (int kk = 0; kk < 4; ++kk) {
    const int k = wave * 4 + kk;
    const float* __restrict__ Rk = P + (size_t)(48 + k) * H;
    float acc = 0.0f;
#pragma unroll 4
    for (int t = 0; t < 32; ++t) {
      const int h = t * 32 + lane;
      float v = sSO[h] + Rk[h];
      v = fmaxf(v, 0.0f);
      acc = fmaf(v, sW2[h], acc);
    }
    // wave32 reduction
#pragma unroll
    for (int off = 16; off > 0; off >>= 1) acc += __shfl_down(acc, off, 32);
    if (lane == 0) {
      const float logit = acc + bias2;
      outp[k] = 1.0f / (1.0f + __expf(-logit));
    }
  }
}

extern "C" void kernel_launch(void* const* d_in, const int* in_sizes, int n_in,
                              void* d_out, int out_size, void* d_ws, size_t ws_size,
                              hipStream_t stream) {
  const float* ent    = (const float*)d_in[0];  // entity_emb [48,1024]
  const float* rel    = (const float*)d_in[1];  // rel_emb    [32,1024]
  const float* W1     = (const float*)d_in[2];  // [3072,1024] row-major
  const float* b1     = (const float*)d_in[3];  // [1024]
  const float* W2     = (const float*)d_in[4];  // [1024,1]
  const float* b2     = (const float*)d_in[5];  // [1]
  const int*   starts = (const int*)d_in[6];    // [48]
  const int*   maxd   = (const int*)d_in[7];    // scalar
  float*       out    = (float*)d_out;          // [48,48,32]
  float*       P      = (float*)d_ws;           // 128*1024 f32 = 512 KB scratch

  relik_stage1_wmma<<<dim3(8, 4), 128, 0, stream>>>(ent, rel, W1, P);
  relik_stage2<<<dim3(48, 48), 256, 0, stream>>>(P, b1, W2, b2, starts, maxd, out);
}